// n_pair_loss_35098472743389
// MI455X (gfx1250) — compile-verified
//
#include <hip/hip_runtime.h>
#include <math.h>

typedef __attribute__((ext_vector_type(2))) float v2f;
typedef __attribute__((ext_vector_type(8))) float v8f;

#define DIM    128
#define NPAIR  8192
#define L2REG  0.05f
#define MBLK   2          // anchor M-tiles per wave (register blocking)
#define JSPLIT 2          // j-range partitions per anchor tile

// ---------------------------------------------------------------------------
// Per-row pass: diag[i] = dot(anc_i, pos_i); accumulate pre-scaled norm terms.
// ---------------------------------------------------------------------------
__global__ __launch_bounds__(256) void aux_kernel(const float* __restrict__ x,
                                                  float* __restrict__ diag,
                                                  float* __restrict__ acc_norm) {
  int i = blockIdx.x * blockDim.x + threadIdx.x;   // pair row 0..8191
  const float4* a = (const float4*)(x + (size_t)(2 * i) * DIM);
  const float4* p = (const float4*)(x + (size_t)(2 * i + 1) * DIM);
  float dot = 0.f, na = 0.f, np2 = 0.f;
#pragma unroll
  for (int k = 0; k < DIM / 4; ++k) {
    float4 av = a[k], pv = p[k];
    dot += av.x * pv.x + av.y * pv.y + av.z * pv.z + av.w * pv.w;
    na  += av.x * av.x + av.y * av.y + av.z * av.z + av.w * av.w;
    np2 += pv.x * pv.x + pv.y * pv.y + pv.z * pv.z + pv.w * pv.w;
  }
  diag[i] = dot;
  float contrib = L2REG * (sqrtf(na) * (1.0f / NPAIR) + sqrtf(np2) * (1.0f / DIM));
#pragma unroll
  for (int off = 16; off > 0; off >>= 1)
    contrib += __shfl_down(contrib, off, 32);
  if ((threadIdx.x & 31) == 0) atomicAdd(acc_norm, contrib);
}

// ---------------------------------------------------------------------------
// Fused GEMM + online logsumexp, M-blocked x2: each wave owns 32 anchor rows
// (two 16x128 A operands resident in 128 VGPRs) and streams pos tiles over
// half the j-range; every B float2 feeds two V_WMMA_F32_16X16X4_F32.
// Emits partial (m, s) per row into workspace.
// ---------------------------------------------------------------------------
__global__ __launch_bounds__(64) void lse_wmma_kernel(const float* __restrict__ x,
                                                      float* __restrict__ mpart,
                                                      float* __restrict__ spart) {
  const int lane = threadIdx.x & 31;
  const int wave = threadIdx.x >> 5;
  const int g    = blockIdx.x * 2 + wave;   // 0..511
  const int tile = g >> 1;                  // 0..255 -> anchor rows tile*32..+31
  const int part = g & 1;                   // which j half
  const int r    = lane & 15;
  const int h    = lane >> 4;

  // A operands: rows tile*32 + mb*16 + r (global even rows), k = 4t + 2h + {0,1}
  v2f a[MBLK][32];
#pragma unroll
  for (int mb = 0; mb < MBLK; ++mb) {
    const float* arow = x + (size_t)(2 * (tile * 32 + mb * 16 + r)) * DIM + 2 * h;
#pragma unroll
    for (int t = 0; t < 32; ++t) a[mb][t] = *(const v2f*)(arow + 4 * t);
  }

  float m[MBLK][8], s[MBLK][8];
#pragma unroll
  for (int mb = 0; mb < MBLK; ++mb)
#pragma unroll
    for (int v = 0; v < 8; ++v) { m[mb][v] = -__builtin_inff(); s[mb][v] = 0.f; }

  const int jbeg = part * (NPAIR / JSPLIT);
  const int jend = jbeg + (NPAIR / JSPLIT);

  for (int j0 = jbeg; j0 < jend; j0 += 16) {
    const float* prow = x + (size_t)(2 * (j0 + r) + 1) * DIM + 2 * h;
    // Prefetch next tile's row one iteration ahead (gfx1250 global_prefetch).
    __builtin_prefetch(prow + 16 * DIM * 2, 0, 3);

    v8f c0 = {0.f, 0.f, 0.f, 0.f, 0.f, 0.f, 0.f, 0.f};
    v8f c1 = {0.f, 0.f, 0.f, 0.f, 0.f, 0.f, 0.f, 0.f};
#pragma unroll
    for (int t = 0; t < 32; ++t) {
      v2f b = *(const v2f*)(prow + 4 * t);
      c0 = __builtin_amdgcn_wmma_f32_16x16x4_f32(false, a[0][t], false, b,
                                                 (short)0, c0, false, false);
      c1 = __builtin_amdgcn_wmma_f32_16x16x4_f32(false, a[1][t], false, b,
                                                 (short)0, c1, false, false);
    }
#pragma unroll
    for (int v = 0; v < 8; ++v) {
      float g0 = c0[v], g1 = c1[v];
      float n0 = fmaxf(m[0][v], g0);
      float n1 = fmaxf(m[1][v], g1);
      s[0][v] = s[0][v] * __expf(m[0][v] - n0) + __expf(g0 - n0);
      s[1][v] = s[1][v] * __expf(m[1][v] - n1) + __expf(g1 - n1);
      m[0][v] = n0;
      m[1][v] = n1;
    }
  }

  // Combine (m,s) across the 16 lanes of each half-wave.
#pragma unroll
  for (int mb = 0; mb < MBLK; ++mb)
#pragma unroll
    for (int v = 0; v < 8; ++v) {
#pragma unroll
      for (int mask = 8; mask >= 1; mask >>= 1) {
        float om = __shfl_xor(m[mb][v], mask, 16);
        float os = __shfl_xor(s[mb][v], mask, 16);
        float nm = fmaxf(m[mb][v], om);
        s[mb][v] = s[mb][v] * __expf(m[mb][v] - nm) + os * __expf(om - nm);
        m[mb][v] = nm;
      }
    }

  if (r == 0) {  // lanes 0 and 16 hold rows v and v+8 of each tile
    size_t base = (size_t)part * NPAIR;
#pragma unroll
    for (int mb = 0; mb < MBLK; ++mb)
#pragma unroll
      for (int v = 0; v < 8; ++v) {
        int row = tile * 32 + mb * 16 + v + 8 * h;
        mpart[base + row] = m[mb][v];
        spart[base + row] = s[mb][v];
      }
  }
}

// ---------------------------------------------------------------------------
// Merge the JSPLIT partial logsumexp states, subtract diagonal, reduce.
// ---------------------------------------------------------------------------
__global__ __launch_bounds__(256) void combine_kernel(const float* __restrict__ mpart,
                                                      const float* __restrict__ spart,
                                                      const float* __restrict__ diag,
                                                      float* __restrict__ acc_lse) {
  int i = blockIdx.x * blockDim.x + threadIdx.x;   // row 0..8191
  float m0 = mpart[i],         s0 = spart[i];
  float m1 = mpart[NPAIR + i], s1 = spart[NPAIR + i];
  float nm = fmaxf(m0, m1);
  float ss = s0 * __expf(m0 - nm) + s1 * __expf(m1 - nm);
  float lse = nm + __logf(ss) - diag[i];
#pragma unroll
  for (int off = 16; off > 0; off >>= 1)
    lse += __shfl_down(lse, off, 32);
  if ((threadIdx.x & 31) == 0) atomicAdd(acc_lse, lse);
}

__global__ void finalize_kernel(const float* __restrict__ acc,
                                float* __restrict__ out) {
  out[0] = acc[0] * (1.0f / NPAIR) + acc[1];
}

extern "C" void kernel_launch(void* const* d_in, const int* in_sizes, int n_in,
                              void* d_out, int out_size, void* d_ws, size_t ws_size,
                              hipStream_t stream) {
  const float* x = (const float*)d_in[0];   // image_embed (16384 x 128) fp32
  // d_in[1] (labels) is unused by the math.
  float* ws    = (float*)d_ws;
  float* acc   = ws;                         // acc[0]=sum(lse-diag), acc[1]=norms
  float* diag  = ws + 2;                     // 8192
  float* mpart = diag + NPAIR;               // 2*8192
  float* spart = mpart + JSPLIT * NPAIR;     // 2*8192

  hipMemsetAsync(ws, 0, 2 * sizeof(float), stream);
  aux_kernel<<<NPAIR / 256, 256, 0, stream>>>(x, diag, acc + 1);
  lse_wmma_kernel<<<256, 64, 0, stream>>>(x, mpart, spart);
  combine_kernel<<<NPAIR / 256, 256, 0, stream>>>(mpart, spart, diag, acc);
  finalize_kernel<<<1, 1, 0, stream>>>(acc, (float*)d_out);
}